// EdgeSIMPLEBatched_31714038513983
// MI455X (gfx1250) — compile-verified
//
#include <hip/hip_runtime.h>
#include <hip/hip_bf16.h>
#include <stdint.h>

// Problem constants (match reference)
#define BSZ  128
#define NMAX 64
#define ENS  4
#define KK   32          // K
#define NN   4096        // N == NSQ (no padding needed)
#define RR   512         // BSZ * ENS
#define NEGV (-1e30f)

typedef uint32_t u32;
typedef __attribute__((ext_vector_type(4))) u32 v4u_t;
typedef __attribute__((ext_vector_type(8))) int v8i_t;
typedef __attribute__((ext_vector_type(4))) int v4i_t;

// ---------------------------------------------------------------------------
// Math helpers (match jnp semantics)
// ---------------------------------------------------------------------------
__device__ __forceinline__ float lae(float a, float b) {
  // logaddexp
  float m = fmaxf(a, b);
  float d = fminf(a, b) - m;
  return m + log1pf(__expf(d));
}

__device__ __forceinline__ float logsig_theta(float s) {
  // theta = -softplus(-s) + 1e-7
  float z  = -s;
  float sp = fmaxf(z, 0.f) + log1pf(__expf(-fabsf(z)));
  return -sp + 1e-7f;
}

__device__ __forceinline__ float log1mexp_f(float x) {
  // stable log(1 - exp(x)), x <= 0
  x = fminf(x, -1e-7f);
  if (x < -0.6931472f) return log1pf(-__expf(x));
  return __logf(-expm1f(x));
}

// ---------------------------------------------------------------------------
// CDNA5 Tensor Data Mover: 1-row tile load (contiguous nfloats) into LDS.
// D# built per CDNA5 ISA (groups 0/1; groups 2/3
// zero => <=2D tensor). workgroup_mask = 0 (not in a cluster).
// This toolchain's builtin is the 6-arg form:
//   (uint32x4 g0, int32x8 g1, int32x4 g2, int32x4 g3, int32x8 extra, i32 cpol)
// ---------------------------------------------------------------------------
#if defined(__has_builtin)
#if __has_builtin(__builtin_amdgcn_tensor_load_to_lds)
#define HAS_TDM 1
#endif
#if __has_builtin(__builtin_amdgcn_s_wait_tensorcnt)
#define WAIT_TENSORCNT(k) __builtin_amdgcn_s_wait_tensorcnt((short)(k))
#endif
#endif
#ifndef WAIT_TENSORCNT
#define WAIT_TENSORCNT(k)
#endif

__device__ __forceinline__ void tdm_load_f32(const float* gsrc, void* ldsdst, int nfloats) {
#ifdef HAS_TDM
  v4u_t g0; v8i_t g1; v4i_t z4; v8i_t z8;
  z4[0] = 0; z4[1] = 0; z4[2] = 0; z4[3] = 0;
  z8[0] = 0; z8[1] = 0; z8[2] = 0; z8[3] = 0;
  z8[4] = 0; z8[5] = 0; z8[6] = 0; z8[7] = 0;
  unsigned long long ga = (unsigned long long)(uintptr_t)gsrc;
  // Flat LDS aperture keeps the byte offset in the low 32 bits (ISA 10.2).
  u32 laddr = (u32)(uintptr_t)ldsdst;
  // group0: count=1 | lds_addr | global_addr[31:0] | global_addr[56:32], type=2
  g0[0] = 1u;
  g0[1] = laddr;
  g0[2] = (u32)(ga & 0xffffffffull);
  g0[3] = (u32)((ga >> 32) & 0x01ffffffull) | 0x80000000u;
  // group1: data_size=4B (code 2); tensor_dim0 = nfloats; tensor_dim1 = 1;
  //         tile_dim0 = nfloats; tile_dim1 = 1; dim0 stride = nfloats.
  g1[0] = 0x00020000;
  g1[1] = (int)(((u32)nfloats & 0xffffu) << 16);                // tensor_dim0 lo16
  g1[2] = (int)((((u32)nfloats >> 16) & 0xffffu) | (1u << 16)); // dim0 hi | dim1 lo
  g1[3] = (int)(((u32)nfloats & 0xffffu) << 16);                // tile_dim0
  g1[4] = 1;                                                    // tile_dim1
  g1[5] = nfloats;                                              // tensor_dim0_stride lo32
  g1[6] = 0;
  g1[7] = 0;
  __builtin_amdgcn_tensor_load_to_lds(g0, g1, z4, z4, z8, 0);
#else
  // Fallback: cooperative copy (single wave per block in the scan kernels).
  float* d = (float*)ldsdst;
  for (int i = (int)threadIdx.x; i < nfloats; i += 32) d[i] = gsrc[i];
#endif
}

// ---------------------------------------------------------------------------
// K1: prep — logp/logq in (R, N) layout from permuted scores.
// scores flat index for (r = b*4+e, n): b*16384 + n*4 + e
// ---------------------------------------------------------------------------
__global__ __launch_bounds__(256)
void simple_prep(const float* __restrict__ scores,
                 float* __restrict__ lp, float* __restrict__ lq) {
  size_t t = (size_t)blockIdx.x * 256 + threadIdx.x;   // t = r*NN + n
  int r = (int)(t / NN);
  int n = (int)(t % NN);
  int b = r >> 2, e = r & 3;
  float s  = scores[(size_t)b * (NMAX * NMAX * ENS) + (size_t)n * ENS + e];
  float th = logsig_theta(s);
  lp[t] = th;
  lq[t] = log1mexp_f(th);
}

// ---------------------------------------------------------------------------
// K2: both scans in ONE kernel so the independent suffix and prefix chains
// run concurrently (1024 waves instead of 2 serialized 512-wave phases).
// One wave per row r; lane j holds DP state j (j=0..31); state 32 is kept
// wave-uniform. TDM double-buffers lp/lq chunks in LDS.
// ---------------------------------------------------------------------------
#define CH 512              // steps per TDM chunk (2 KB per array per buffer)
#define NCHUNK (NN / CH)    // 8

__device__ __forceinline__ void suffix_scan_row(
    int r, int lane, const float* __restrict__ lp, const float* __restrict__ lq,
    float* __restrict__ logS, float (*bufP)[CH], float (*bufQ)[CH]) {
  const float* lpr = lp + (size_t)r * NN;
  const float* lqr = lq + (size_t)r * NN;
  float* Sr = logS + (size_t)r * (NN + 1) * 33;

  float st  = (lane == 0) ? 0.f : NEGV;  // Pr(count=j) = 1 at j=0
  float s32 = NEGV;
  Sr[(size_t)NN * 33 + lane] = st;       // logS[N] = init
  if (lane == 31) Sr[(size_t)NN * 33 + 32] = s32;

  tdm_load_f32(lpr + (NN - CH), &bufP[0][0], CH);
  tdm_load_f32(lqr + (NN - CH), &bufQ[0][0], CH);

  for (int c = 0; c < NCHUNK; ++c) {
    const int buf = c & 1;
    const int nlo = NN - (c + 1) * CH;
    if (c + 1 < NCHUNK) {
      tdm_load_f32(lpr + (nlo - CH), &bufP[buf ^ 1][0], CH);
      tdm_load_f32(lqr + (nlo - CH), &bufQ[buf ^ 1][0], CH);
      WAIT_TENSORCNT(2);     // current buffer's two loads complete (in-order)
    } else {
      WAIT_TENSORCNT(0);
    }
    for (int t = CH - 1; t >= 0; --t) {
      const int n = nlo + t;
      float lpv = bufP[buf][t];
      float lqv = bufQ[buf][t];
      float up  = __shfl_up(st, 1, 32);      // state[j-1]
      if (lane == 0) up = NEGV;
      float old31 = __shfl(st, 31, 32);      // pre-update state[31], uniform
      s32 = lae(lpv + old31, lqv + s32);
      st  = lae(lpv + up,    lqv + st);
      const size_t base = (size_t)n * 33;
      Sr[base + lane] = st;
      if (lane == 31) Sr[base + 32] = s32;
    }
    asm volatile("s_wait_dscnt 0" ::: "memory");  // LDS reads done before TDM overwrite
  }
}

__device__ __forceinline__ void prefix_scan_row(
    int r, int lane, const float* __restrict__ lp, const float* __restrict__ lq,
    float* __restrict__ logF, float (*bufP)[CH], float (*bufQ)[CH]) {
  const float* lpr = lp + (size_t)r * NN;
  const float* lqr = lq + (size_t)r * NN;
  float* Fr = logF + (size_t)r * NN * 33;

  float st  = (lane == 0) ? 0.f : NEGV;
  float s32 = NEGV;

  tdm_load_f32(lpr, &bufP[0][0], CH);
  tdm_load_f32(lqr, &bufQ[0][0], CH);

  for (int c = 0; c < NCHUNK; ++c) {
    const int buf = c & 1;
    const int nlo = c * CH;
    if (c + 1 < NCHUNK) {
      tdm_load_f32(lpr + (nlo + CH), &bufP[buf ^ 1][0], CH);
      tdm_load_f32(lqr + (nlo + CH), &bufQ[buf ^ 1][0], CH);
      WAIT_TENSORCNT(2);
    } else {
      WAIT_TENSORCNT(0);
    }
    for (int t = 0; t < CH; ++t) {
      const int n = nlo + t;
      const size_t base = (size_t)n * 33;
      // logF is written once and read once -> non-temporal (don't pollute L2)
      __builtin_nontemporal_store(st, &Fr[base + lane]);
      if (lane == 31) __builtin_nontemporal_store(s32, &Fr[base + 32]);
      float lpv = bufP[buf][t];
      float lqv = bufQ[buf][t];
      float up  = __shfl_up(st, 1, 32);
      if (lane == 0) up = NEGV;
      float old31 = __shfl(st, 31, 32);
      s32 = lae(lpv + old31, lqv + s32);
      st  = lae(lpv + up,    lqv + st);
    }
    asm volatile("s_wait_dscnt 0" ::: "memory");
  }
}

__global__ __launch_bounds__(32)
void simple_scan_both(const float* __restrict__ lp, const float* __restrict__ lq,
                      float* __restrict__ logS, float* __restrict__ logF) {
  __shared__ float bufP[2][CH];
  __shared__ float bufQ[2][CH];
  const int lane = threadIdx.x;
  if (blockIdx.x < RR) {
    suffix_scan_row((int)blockIdx.x, lane, lp, lq, logS, bufP, bufQ);
  } else {
    prefix_scan_row((int)blockIdx.x - RR, lane, lp, lq, logF, bufP, bufQ);
  }
}

// ---------------------------------------------------------------------------
// K3: exact marginals, one thread per (r, n):
// logmu = logp + lse_{j=0..K-1}(F[n][j] + S[n+1][K-1-j]) - logS[0][K]
// ---------------------------------------------------------------------------
__global__ __launch_bounds__(256)
void simple_marginal(const float* __restrict__ lp,
                     const float* __restrict__ logS,
                     const float* __restrict__ logF,
                     float* __restrict__ mu) {
  size_t t = (size_t)blockIdx.x * 256 + threadIdx.x;   // t = r*NN + n
  int r = (int)(t / NN);
  int n = (int)(t % NN);
  const float* F  = logF + ((size_t)r * NN + n) * 33;
  const float* S1 = logS + ((size_t)r * (NN + 1) + (size_t)(n + 1)) * 33;
  float logS0K = logS[(size_t)r * (NN + 1) * 33 + KK];

  float term[KK];
  float m = NEGV;
#pragma unroll
  for (int j = 0; j < KK; ++j) {
    term[j] = __builtin_nontemporal_load(&F[j]) + S1[KK - 1 - j];
    m = fmaxf(m, term[j]);
  }
  float sum = 0.f;
#pragma unroll
  for (int j = 0; j < KK; ++j) sum += __expf(term[j] - m);

  float lpv   = lp[t];
  float logmu = lpv + m + __logf(sum) - logS0K;
  mu[t] = __expf(logmu);
}

// ---------------------------------------------------------------------------
// K4: exact conditional-Bernoulli sampling + straight-through combine.
// One wave per row: budget counter rc is wave-uniform, so the data-dependent
// logS lookups become __shfl broadcasts. Writes the permuted output directly.
// ---------------------------------------------------------------------------
__global__ __launch_bounds__(32)
void simple_sample(const float* __restrict__ lp,
                   const float* __restrict__ unif,
                   const float* __restrict__ mu,
                   const float* __restrict__ logS,
                   float* __restrict__ out) {
  const int r    = blockIdx.x;
  const int lane = threadIdx.x;
  const float* Sr  = logS + (size_t)r * (NN + 1) * 33;
  const float* lpr = lp   + (size_t)r * NN;
  const float* ur  = unif + (size_t)r * NN;   // unif is (R, N); step n uses unif[r, n]
  const float* mur = mu   + (size_t)r * NN;
  const int b = r >> 2, e = r & 3;
  float* outb = out + (size_t)b * (NMAX * NMAX * ENS) + e;

  int rc = KK;
  float cur   = Sr[lane];   // logS[0][lane]
  float cur32 = Sr[KK];     // logS[0][32] (uniform)

  for (int n = 0; n < NN; ++n) {
    const float* Snx = Sr + (size_t)(n + 1) * 33;
    if (n + 8 <= NN) __builtin_prefetch(Sr + (size_t)(n + 8) * 33, 0, 0);
    float nxt   = Snx[lane];
    float nxt32 = Snx[KK];
    float lpv = lpr[n];
    float u   = ur[n];
    float muv = mur[n];
    int i1 = rc - 1; i1 = (i1 < 0) ? 0 : ((i1 > KK) ? KK : i1);
    int i0 = rc;     i0 = (i0 < 0) ? 0 : ((i0 > KK) ? KK : i0);
    float num = lpv + ((i1 == KK) ? nxt32 : __shfl(nxt, i1, 32));
    float den =        (i0 == KK) ? cur32 : __shfl(cur, i0, 32);
    float p1 = __expf(fminf(num - den, 0.f));
    int x = (u < p1 && rc > 0) ? 1 : 0;
    rc -= x;
    float o = ((float)x + muv) - muv;   // straight-through, reference FP order
    if (lane == 0) outb[(size_t)n * ENS] = o;
    cur = nxt; cur32 = nxt32;
  }
}

// ---------------------------------------------------------------------------
// Host launcher
// ---------------------------------------------------------------------------
extern "C" void kernel_launch(void* const* d_in, const int* in_sizes, int n_in,
                              void* d_out, int out_size, void* d_ws, size_t ws_size,
                              hipStream_t stream) {
  const float* scores = (const float*)d_in[0];   // (BSZ, NMAX, NMAX, ENS) f32
  const float* unif   = (const float*)d_in[1];   // (R, N) f32
  float* out = (float*)d_out;                    // (BSZ, NMAX, NMAX, ENS) f32

  char* ws = (char*)d_ws;
  const size_t SZ_PQ   = (size_t)RR * NN * sizeof(float);              // 8 MB each
  const size_t SZ_LOGS = (size_t)RR * (NN + 1) * 33 * sizeof(float);   // ~277 MB
  const size_t SZ_LOGF = (size_t)RR * NN * 33 * sizeof(float);         // ~277 MB
  float* lp   = (float*)(ws);
  float* lq   = (float*)(ws + SZ_PQ);
  float* logS = (float*)(ws + 2 * SZ_PQ);
  float* logF = (float*)(ws + 2 * SZ_PQ + SZ_LOGS);
  float* mu   = (float*)(ws + 2 * SZ_PQ + SZ_LOGS + SZ_LOGF);

  simple_prep     <<<(RR * NN) / 256, 256, 0, stream>>>(scores, lp, lq);
  simple_scan_both<<<2 * RR, 32, 0, stream>>>(lp, lq, logS, logF);
  simple_marginal <<<(RR * NN) / 256, 256, 0, stream>>>(lp, logS, logF, mu);
  simple_sample   <<<RR, 32, 0, stream>>>(lp, unif, mu, logS, out);
}